// WeakREChead_47923245088965
// MI455X (gfx1250) — compile-verified
//
#include <hip/hip_runtime.h>
#include <math.h>

typedef __attribute__((ext_vector_type(2))) float v2f;
typedef __attribute__((ext_vector_type(4))) float v4f;
typedef __attribute__((ext_vector_type(8))) float v8f;

#define BB 128
#define VV 400
#define DD 768
#define NN (BB * VV) /* 51200 */

#define KC 96          /* d-chunk staged in LDS */
#define PKC (KC + 4)   /* padded row stride (floats): bank-conflict-free */

// ---------------------------------------------------------------------------
// Kernel 1: S[b, a*400+v] = sum_d (fusion[a,v,d]+tag[a,v,d]) * lan[b,d]
// Block = 4 waves; each wave owns one 16-wide N tile and all 128 b rows
// (8 WMMA accumulator tiles). lan chunk [128 x KC] staged in LDS once per
// block (shared by 4 waves); comb streams from HBM with prefetch.
// ---------------------------------------------------------------------------
__global__ __launch_bounds__(128) void gemm_s_kernel(
    const float* __restrict__ fusion, const float* __restrict__ tag,
    const float* __restrict__ lan, float* __restrict__ S) {
  __shared__ float slan[BB * PKC];  // 128 * 100 * 4B = 50 KB

  const int tid  = threadIdx.x;
  const int wave = tid >> 5;
  const int tile = blockIdx.x * 4 + wave;  // 0..3199
  const int lane = tid & 31;
  const int half = lane >> 4;  // K pair {0,1} vs {2,3}
  const int l16  = lane & 15;  // B-frag column / A-frag row
  const int n0   = tile * 16;

  const float* frow = fusion + (size_t)(n0 + l16) * DD + half * 2;
  const float* trow = tag    + (size_t)(n0 + l16) * DD + half * 2;

  v8f acc[8];
#pragma unroll
  for (int t = 0; t < 8; ++t) acc[t] = (v8f){};

  for (int d0 = 0; d0 < DD; d0 += KC) {
    // ---- stage lan[0:128, d0:d0+KC] into padded LDS (coalesced b128) ----
    for (int s = tid; s < BB * (KC / 4); s += 128) {
      int row = s / (KC / 4);
      int c4  = s - row * (KC / 4);
      v4f val = *(const v4f*)(lan + (size_t)row * DD + d0 + c4 * 4);
      *(v4f*)(slan + row * PKC + c4 * 4) = val;
    }
    __syncthreads();

    // prefetch next chunk of comb rows into cache hierarchy
    if (d0 + KC < DD) {
      __builtin_prefetch(frow + d0 + KC, 0, 1);
      __builtin_prefetch(trow + d0 + KC, 0, 1);
    }

    // ---- K loop over this chunk: 8 WMMAs per k-step ----
    for (int dk = 0; dk < KC; dk += 4) {
      v2f bf = *(const v2f*)(frow + d0 + dk) + *(const v2f*)(trow + d0 + dk);
      const float* sl = slan + l16 * PKC + dk + half * 2;
#pragma unroll
      for (int t = 0; t < 8; ++t) {
        v2f af = *(const v2f*)(sl + t * 16 * PKC);
        acc[t] = __builtin_amdgcn_wmma_f32_16x16x4_f32(
            false, af, false, bf, (short)0, acc[t], false, false);
      }
    }
    __syncthreads();
  }

  // C layout: VGPR r, lanes 0-15 -> M=r, N=lane; lanes 16-31 -> M=r+8
#pragma unroll
  for (int t = 0; t < 8; ++t) {
#pragma unroll
    for (int r = 0; r < 8; ++r) {
      int b = t * 16 + r + half * 8;
      S[(size_t)b * NN + n0 + l16] = acc[t][r];
    }
  }
}

// ---------------------------------------------------------------------------
// Kernel 2: per-(b,a) top-2 over v (400 values)
// ---------------------------------------------------------------------------
__global__ __launch_bounds__(128) void top2_kernel(
    const float* __restrict__ S, float* __restrict__ max1,
    float* __restrict__ max2) {
  const int aa = blockIdx.x, bb = blockIdx.y;
  const float* row = S + (size_t)bb * NN + (size_t)aa * VV;
  const int t = threadIdx.x;
  float m1 = -__builtin_inff(), m2 = -__builtin_inff();
  for (int v = t; v < VV; v += 128) {
    float x = row[v];
    if (x > m1) { m2 = m1; m1 = x; } else if (x > m2) { m2 = x; }
  }
  __shared__ float s1[128], s2[128];
  s1[t] = m1; s2[t] = m2;
  __syncthreads();
  for (int off = 64; off > 0; off >>= 1) {
    if (t < off) {
      float a1 = s1[t], a2 = s2[t], o1 = s1[t + off], o2 = s2[t + off];
      float n1, n2;
      if (a1 >= o1) { n1 = a1; n2 = fmaxf(o1, a2); }
      else          { n1 = o1; n2 = fmaxf(a1, o2); }
      s1[t] = n1; s2[t] = n2;
    }
    __syncthreads();
  }
  if (t == 0) {
    max1[(size_t)bb * BB + aa] = s1[0];
    max2[(size_t)bb * BB + aa] = s2[0];
  }
}

// ---------------------------------------------------------------------------
// Kernel 3: Fsum[v,d] = sum_a fusion[a,v,d]
// ---------------------------------------------------------------------------
__global__ void fsum_kernel(const float* __restrict__ fusion,
                            float* __restrict__ Fsum) {
  int idx = blockIdx.x * blockDim.x + threadIdx.x;
  if (idx >= VV * DD) return;
  float s = 0.0f;
  for (int a = 0; a < BB; ++a) s += fusion[(size_t)a * VV * DD + idx];
  Fsum[idx] = s;
}

// ---------------------------------------------------------------------------
// Kernel 4: sel[b,v] = dot(Fsum[v,:], lan[b,:]); first-argmax_v; gather
// fusion[b, idx, :] -> out_emb. One block per b; one wave per v (coalesced
// lane-split dot + shfl_xor wave32 reduction).
// ---------------------------------------------------------------------------
__global__ __launch_bounds__(256) void anchor_kernel(
    const float* __restrict__ fusion, const float* __restrict__ lan,
    const float* __restrict__ Fsum, float* __restrict__ out_emb) {
  const int b = blockIdx.x;
  __shared__ float sel[VV];
  __shared__ int s_idx;
  const float* lrow = lan + (size_t)b * DD;
  const int w = threadIdx.x >> 5, lane = threadIdx.x & 31;
  for (int v = w; v < VV; v += 8) {
    const float* fr = Fsum + (size_t)v * DD;
    float acc = 0.0f;
    for (int d = lane; d < DD; d += 32) acc += fr[d] * lrow[d];
#pragma unroll
    for (int m = 16; m > 0; m >>= 1) acc += __shfl_xor(acc, m, 32);
    if (lane == 0) sel[v] = acc;
  }
  __syncthreads();
  if (threadIdx.x == 0) {
    int best = 0; float bv = sel[0];
    for (int v = 1; v < VV; ++v)
      if (sel[v] > bv) { bv = sel[v]; best = v; }
    s_idx = best;
  }
  __syncthreads();
  const float* src = fusion + ((size_t)b * VV + s_idx) * DD;
  for (int d = threadIdx.x; d < DD; d += blockDim.x)
    out_emb[(size_t)b * DD + d] = src[d];
}

// ---------------------------------------------------------------------------
// Kernel 5: logits = [max1 row (128) | max2 row off-diag (127)] -> log_softmax
// -> loss = -mean_b logp[b, b]
// ---------------------------------------------------------------------------
__global__ __launch_bounds__(128) void loss_kernel(
    const float* __restrict__ max1, const float* __restrict__ max2,
    float* __restrict__ out_loss) {
  const int b = threadIdx.x;
  const float* r1 = max1 + (size_t)b * BB;
  const float* r2 = max2 + (size_t)b * BB;
  float rowmax = -__builtin_inff();
  for (int j = 0; j < BB; ++j) rowmax = fmaxf(rowmax, r1[j]);
  for (int j = 0; j < BB; ++j) if (j != b) rowmax = fmaxf(rowmax, r2[j]);
  float se = 0.0f;
  for (int j = 0; j < BB; ++j) se += expf(r1[j] - rowmax);
  for (int j = 0; j < BB; ++j) if (j != b) se += expf(r2[j] - rowmax);
  float lp = (r1[b] - rowmax) - logf(se);
  __shared__ float red[128];
  red[b] = -lp;
  __syncthreads();
  for (int off = 64; off > 0; off >>= 1) {
    if (b < off) red[b] += red[b + off];
    __syncthreads();
  }
  if (b == 0) out_loss[0] = red[0] / (float)BB;
}

// ---------------------------------------------------------------------------
// Kernel 6: predictions = one_hot(argmax_v S[b, b*400+v]) as float 0/1
// ---------------------------------------------------------------------------
__global__ __launch_bounds__(128) void pred_kernel(
    const float* __restrict__ S, float* __restrict__ out_pred) {
  const int b = blockIdx.x;
  const float* row = S + (size_t)b * NN + (size_t)b * VV;
  __shared__ int s_idx;
  if (threadIdx.x == 0) {
    int best = 0; float bv = row[0];
    for (int v = 1; v < VV; ++v)
      if (row[v] > bv) { bv = row[v]; best = v; }
    s_idx = best;
  }
  __syncthreads();
  for (int v = threadIdx.x; v < VV; v += blockDim.x)
    out_pred[(size_t)b * VV + v] = (v == s_idx) ? 1.0f : 0.0f;
}

extern "C" void kernel_launch(void* const* d_in, const int* in_sizes, int n_in,
                              void* d_out, int out_size, void* d_ws,
                              size_t ws_size, hipStream_t stream) {
  const float* fusion = (const float*)d_in[0];
  const float* lan    = (const float*)d_in[1];
  const float* tag    = (const float*)d_in[2];
  float* out = (float*)d_out;

  // workspace layout (floats): S | max1 | max2 | Fsum   (~27.5 MB)
  float* S    = (float*)d_ws;
  float* max1 = S + (size_t)BB * NN;
  float* max2 = max1 + BB * BB;
  float* Fsum = max2 + BB * BB;

  float* out_loss = out;                 // 1
  float* out_emb  = out + 1;             // 128*768
  float* out_pred = out + 1 + BB * DD;   // 128*400

  gemm_s_kernel<<<NN / 16 / 4, 128, 0, stream>>>(fusion, tag, lan, S);
  top2_kernel<<<dim3(BB, BB), 128, 0, stream>>>(S, max1, max2);
  fsum_kernel<<<(VV * DD + 255) / 256, 256, 0, stream>>>(fusion, Fsum);
  anchor_kernel<<<BB, 256, 0, stream>>>(fusion, lan, Fsum, out_emb);
  loss_kernel<<<1, 128, 0, stream>>>(max1, max2, out_loss);
  pred_kernel<<<BB, 128, 0, stream>>>(S, out_pred);
}